// HashEmbedder_36283883717062
// MI455X (gfx1250) — compile-verified
//
#include <hip/hip_runtime.h>
#include <math.h>
#include <stdint.h>

#define NLVL   16
#define LOG2T  19
#define TSIZE  (1u << LOG2T)
#define HMASK  (TSIZE - 1u)
#define PRIME2 2654435761u
#define PRIME3 805459861u
#define BLOCK  256
#define LDS_STRIDE 36   // 32 feature floats + 4 pad -> 8B-aligned b64 writes,
                        // 16B-aligned b128 async-DMA reads

struct LevelParams {
    float r[NLVL];   // resolution
    float g[NLVL];   // grid = 1/r (f32, matches reference's grid value)
};

__global__ __launch_bounds__(BLOCK) void hashgrid_fwd(
    const float* __restrict__ xs,
    const float* __restrict__ tables,
    float* __restrict__ out,
    int n, LevelParams lp)
{
    __shared__ float lds[BLOCK * LDS_STRIDE];

    const int tid = (int)threadIdx.x;
    const int p0  = (int)blockIdx.x * BLOCK;
    const int p   = p0 + tid;
    const int pc  = (p < n) ? p : (n - 1);   // clamp for tail safety

    float px = xs[(size_t)pc * 3 + 0];
    float py = xs[(size_t)pc * 3 + 1];
    float pz = xs[(size_t)pc * 3 + 2];
    px = fminf(fmaxf(px, 0.0f), 1.0f);
    py = fminf(fmaxf(py, 0.0f), 1.0f);
    pz = fminf(fmaxf(pz, 0.0f), 1.0f);

#pragma unroll
    for (int l = 0; l < NLVL; ++l) {
        const float r = lp.r[l];
        const float g = lp.g[l];
        // floor(x/grid) == floor(x*r) up to 1 ulp; trilinear interp is
        // continuous across cell boundaries so any flip is ulp-harmless.
        const float fx = floorf(px * r);
        const float fy = floorf(py * r);
        const float fz = floorf(pz * r);
        const float rx = (px - fx * g) * r;      // rel in [0,1)
        const float ry = (py - fy * g) * r;
        const float rz = (pz - fz * g) * r;

        // hash factors: prime for x is 1
        const uint32_t a0 = (uint32_t)(int)fx;     // <= 513 < 2^19: no mask needed
        const uint32_t a1 = a0 + 1u;
        const uint32_t b0 = (uint32_t)(int)fy * PRIME2;
        const uint32_t b1 = b0 + PRIME2;
        const uint32_t c0 = (uint32_t)(int)fz * PRIME3;
        const uint32_t c1 = c0 + PRIME3;

        // fold the mask into the yz-combination: (a^b^c)&M == a ^ ((b^c)&M)
        const uint32_t d00 = (b0 ^ c0) & HMASK;
        const uint32_t d01 = (b0 ^ c1) & HMASK;
        const uint32_t d10 = (b1 ^ c0) & HMASK;
        const uint32_t d11 = (b1 ^ c1) & HMASK;

        const float2* __restrict__ tb =
            (const float2*)(tables + (size_t)l * TSIZE * 2u);

        // issue all 8 gathers back-to-back (L2-resident, latency-hidden)
        const float2 e000 = tb[a0 ^ d00];
        const float2 e100 = tb[a1 ^ d00];
        const float2 e010 = tb[a0 ^ d10];
        const float2 e110 = tb[a1 ^ d10];
        const float2 e001 = tb[a0 ^ d01];
        const float2 e101 = tb[a1 ^ d01];
        const float2 e011 = tb[a0 ^ d11];
        const float2 e111 = tb[a1 ^ d11];

        const float wx0 = 1.0f - rx;
        const float wy0 = 1.0f - ry;
        const float wz0 = 1.0f - rz;

        const float w00 = wx0 * wy0;   // (dx=0,dy=0)
        const float w01 = wx0 * ry;    // (dx=0,dy=1)
        const float w10 = rx  * wy0;   // (dx=1,dy=0)
        const float w11 = rx  * ry;    // (dx=1,dy=1)

        // z-factored bilinear sums: 4x (mul + 3 fma), then mul+fma combine
        const float s0x = w00 * e000.x + w01 * e010.x + w10 * e100.x + w11 * e110.x;
        const float s1x = w00 * e001.x + w01 * e011.x + w10 * e101.x + w11 * e111.x;
        const float s0y = w00 * e000.y + w01 * e010.y + w10 * e100.y + w11 * e110.y;
        const float s1y = w00 * e001.y + w01 * e011.y + w10 * e101.y + w11 * e111.y;

        float2 o2;
        o2.x = wz0 * s0x + rz * s1x;
        o2.y = wz0 * s0y + rz * s1y;

        // 8B-aligned packed write -> single ds_store_b64
        *(float2*)&lds[tid * LDS_STRIDE + 2 * l] = o2;
    }

    __syncthreads();

    if (p0 + BLOCK <= n) {
        // Coalesced async DMA: LDS -> global. Each lane moves 16B; each wave op
        // covers 512 contiguous bytes of the block's 32KB output tile.
        float* outb = out + (size_t)p0 * 32u;
#pragma unroll
        for (int rr = 0; rr < 8; ++rr) {
            const int c  = rr * BLOCK + tid;   // float4 chunk id in [0, 2048)
            const int pp = c >> 3;             // point within block
            const int j  = (c & 7) << 2;       // feature start (0,4,...,28)
            const uint32_t ldsaddr = (uint32_t)((pp * LDS_STRIDE + j) * 4);
            const uint64_t gaddr   = (uint64_t)(uintptr_t)(outb + (size_t)c * 4u);
            asm volatile("global_store_async_from_lds_b128 %0, %1, off"
                         :: "v"(gaddr), "v"(ldsaddr) : "memory");
        }
        asm volatile("s_wait_asynccnt 0x0" ::: "memory");
    } else {
        // tail block: plain guarded stores
        if (p < n) {
#pragma unroll
            for (int j = 0; j < 32; ++j)
                out[(size_t)p * 32u + j] = lds[tid * LDS_STRIDE + j];
        }
    }
}

extern "C" void kernel_launch(void* const* d_in, const int* in_sizes, int n_in,
                              void* d_out, int out_size, void* d_ws, size_t ws_size,
                              hipStream_t stream) {
    (void)n_in; (void)out_size; (void)d_ws; (void)ws_size;
    const float* x      = (const float*)d_in[0];
    const float* tables = (const float*)d_in[1];
    float* out          = (float*)d_out;
    const int n = in_sizes[0] / 3;

    // Replicate reference's double-precision resolution schedule exactly:
    // b = exp((ln 512 - ln 16)/15); res_i = floor(16 * b^i); grid_i = 1/res_i (f32)
    LevelParams lp;
    const double b = exp((log(512.0) - log(16.0)) / 15.0);
    for (int i = 0; i < NLVL; ++i) {
        lp.r[i] = (float)floor(16.0 * pow(b, (double)i));
        lp.g[i] = 1.0f / lp.r[i];
    }

    const int blocks = (n + BLOCK - 1) / BLOCK;
    hashgrid_fwd<<<blocks, BLOCK, 0, stream>>>(x, tables, out, n, lp);
}